// Encoder_23124103922122
// MI455X (gfx1250) — compile-verified
//
#include <hip/hip_runtime.h>

#define NG      4096   // genera
#define NR      6      // ranks
#define NGRP    256    // groups per rank
#define THREADS 256

// One block per batch row b.
// Phase 1: async-copy F[b,:] (16KB) global -> LDS   (ASYNCcnt path, CDNA5)
// Phase 2: zero 6x256 segment accumulators in LDS
// Phase 3: ds_add_f32 segment-sum; cache labels as packed u8 in LDS
// Phase 4: gather seg[label], emit aligned b128 stores (4 elems = 112B/thread)
__global__ __launch_bounds__(THREADS) void taxa_segsum_kernel(
    const float* __restrict__ F,
    const int*   __restrict__ labels,
    float*       __restrict__ out)
{
    __shared__ float         sF[NG];            // 16384 B
    __shared__ float         seg[NR * NGRP];    //  6144 B
    __shared__ unsigned char sLab[NR * NG];     // 24576 B  (47KB total)

    const int b = blockIdx.x;
    const int t = threadIdx.x;

    // ---- Phase 1: async global->LDS copy of this batch row (16B/lane/iter) ----
    const float* Frow = F + (size_t)b * NG;                  // uniform -> SGPR pair
    unsigned ldsF = (unsigned)(unsigned long long)(void*)sF; // LDS byte offset
    #pragma unroll
    for (int k = 0; k < NG / (THREADS * 4); ++k) {           // 4 iterations
        unsigned byte_off = (unsigned)(t + k * THREADS) * 16u;
        unsigned lds_addr = ldsF + byte_off;
        asm volatile("global_load_async_to_lds_b128 %0, %1, %2"
                     :: "v"(lds_addr), "v"(byte_off), "s"(Frow)
                     : "memory");
    }

    // ---- Phase 2: zero accumulators (independent of async writes) ----
    #pragma unroll
    for (int r = 0; r < NR; ++r)
        seg[r * NGRP + t] = 0.0f;

    asm volatile("s_wait_asynccnt 0" ::: "memory");
    __syncthreads();

    // ---- Phase 3: LDS-atomic segment-sum; stash labels as packed u8 ----
    #pragma unroll
    for (int k = 0; k < NG / (THREADS * 4); ++k) {           // 4 iterations
        int i0 = (t + k * THREADS) * 4;
        float4 f4 = *(const float4*)&sF[i0];
        #pragma unroll
        for (int r = 0; r < NR; ++r) {
            int4 l4 = *(const int4*)&labels[r * NG + i0];    // coalesced 512B/wave
            atomicAdd(&seg[r * NGRP + l4.x], f4.x);          // ds_add_f32
            atomicAdd(&seg[r * NGRP + l4.y], f4.y);
            atomicAdd(&seg[r * NGRP + l4.z], f4.z);
            atomicAdd(&seg[r * NGRP + l4.w], f4.w);
            unsigned packed = (unsigned)l4.x | ((unsigned)l4.y << 8) |
                              ((unsigned)l4.z << 16) | ((unsigned)l4.w << 24);
            *(unsigned*)&sLab[r * NG + i0] = packed;         // conflict-free u32 store
        }
    }
    __syncthreads();

    // ---- Phase 4: gather; 4 consecutive elements -> 28 floats -> 7x b128 ----
    #pragma unroll
    for (int k = 0; k < NG / (THREADS * 4); ++k) {           // 4 iterations
        int i0 = (t + k * THREADS) * 4;
        float4 f4 = *(const float4*)&sF[i0];

        alignas(16) float vals[4 * (NR + 1)];                // 28 regs
        #pragma unroll
        for (int r = 0; r < NR; ++r) {
            unsigned packed = *(const unsigned*)&sLab[r * NG + i0]; // one b32/rank
            vals[0 * (NR + 1) + r] = seg[r * NGRP + ( packed        & 0xffu)];
            vals[1 * (NR + 1) + r] = seg[r * NGRP + ((packed >>  8) & 0xffu)];
            vals[2 * (NR + 1) + r] = seg[r * NGRP + ((packed >> 16) & 0xffu)];
            vals[3 * (NR + 1) + r] = seg[r * NGRP + ( packed >> 24        )];
        }
        vals[0 * (NR + 1) + NR] = f4.x;                      // identity channel
        vals[1 * (NR + 1) + NR] = f4.y;
        vals[2 * (NR + 1) + NR] = f4.z;
        vals[3 * (NR + 1) + NR] = f4.w;

        // out byte offset = ((b*NG + i0)*7)*4 = 112*(b*NG/4 + i0/4) -> 16B aligned
        float4* dst = (float4*)(out + ((size_t)b * NG + i0) * (size_t)(NR + 1));
        #pragma unroll
        for (int w = 0; w < (4 * (NR + 1)) / 4; ++w)         // 7 aligned b128 stores
            dst[w] = *(const float4*)&vals[w * 4];
    }
}

extern "C" void kernel_launch(void* const* d_in, const int* in_sizes, int n_in,
                              void* d_out, int out_size, void* d_ws, size_t ws_size,
                              hipStream_t stream)
{
    const float* F      = (const float*)d_in[0];   // [B, 4096] f32
    const int*   labels = (const int*)d_in[1];     // [6, 4096] i32
    float*       out    = (float*)d_out;           // [B, 4096, 7] f32

    const int B = in_sizes[0] / NG;                // 1024
    hipLaunchKernelGGL(taxa_segsum_kernel, dim3(B), dim3(THREADS), 0, stream,
                       F, labels, out);
}